// NextDist_38379827757416
// MI455X (gfx1250) — compile-verified
//
#include <hip/hip_runtime.h>
#include <hip/hip_bf16.h>
#include <math.h>

// ---------------------------------------------------------------------------
// Problem constants (from the reference)
// ---------------------------------------------------------------------------
#define NG     64          // graphs
#define ATOMS  32          // atoms per graph
#define NN     (NG*ATOMS)  // 2048 nodes
#define HID    128
#define FEAT   64
#define D1BINS 32
#define D2BINS 64
#define GAMMA  10.0f
#define DIM1   106         // dense dims: 128 -> 106 -> 85 -> 64
#define DIM2   85
#define DIM3   64
#define DIM1P  112         // padded to multiple of 16
#define DIM2P  96
#define INV_SQRT_LEN 0.17677669529663687f   // 1/sqrt(32); every graph has 32 atoms

typedef __attribute__((ext_vector_type(2))) float v2f;
typedef __attribute__((ext_vector_type(8))) float v8f;

// fp32 WMMA: D(16x16) = A(16x4) * B(4x16) + C   -- v_wmma_f32_16x16x4_f32
__device__ __forceinline__ v8f wmma_f32_16x16x4(v2f a, v2f b, v8f c) {
#if defined(__HIP_DEVICE_COMPILE__)
    return __builtin_amdgcn_wmma_f32_16x16x4_f32(
        /*neg_a=*/false, a, /*neg_b=*/false, b,
        /*c_mod=*/(short)0, c, /*reuse_a=*/false, /*reuse_b=*/false);
#else
    (void)a; (void)b; return c;
#endif
}

// ---------------------------------------------------------------------------
// Branch-free fp32 WMMA GEMM:  Y[M,N] = X[M,K] @ W[N,K]^T + bias[N]
// Row-major; strides == logical sizes.  REQUIRES: M%16==0, N%16==0, K%4==0
// (callers pad W/bias/activations so this always holds).
// One 16x16 output tile per wave, 8 waves (256 thr) per block.
// K is a compile-time constant; ACT: 0 = none, 1 = SiLU.
// ---------------------------------------------------------------------------
template <int K, int ACT>
__global__ void gemm_wmma_kernel(const float* __restrict__ X,
                                 const float* __restrict__ W,
                                 const float* __restrict__ bias,
                                 float* __restrict__ Y,
                                 int M, int N) {
    const int wave = threadIdx.x >> 5;
    const int lane = threadIdx.x & 31;
    const int tilesN = N >> 4;
    const int tiles  = (M >> 4) * tilesN;
    const int tile   = blockIdx.x * 8 + wave;        // wave-uniform
    if (tile >= tiles) return;                        // whole-wave exit

    const int tm = tile / tilesN;
    const int tn = tile % tilesN;

    const int row   = tm * 16 + (lane & 15);          // A-matrix row
    const int col   = tn * 16 + (lane & 15);          // B/C/D column
    const int khalf = (lane >> 4) * 2;                // K sub-index base per lane half

    const float* __restrict__ xrow = X + (long)row * K + khalf;
    const float* __restrict__ wrow = W + (long)col * K + khalf;

    v8f acc = {0.f,0.f,0.f,0.f,0.f,0.f,0.f,0.f};
    #pragma unroll 8
    for (int kk = 0; kk < K; kk += 4) {
        v2f a, b;
        a.x = xrow[kk];  a.y = xrow[kk + 1];          // contiguous pair -> b64 load
        b.x = wrow[kk];  b.y = wrow[kk + 1];
        acc = wmma_f32_16x16x4(a, b, acc);
    }

    const float bv = bias ? bias[col] : 0.f;
    const int mbase = tm * 16 + 8 * (lane >> 4);
    #pragma unroll
    for (int r = 0; r < 8; ++r) {
        float v = acc[r] + bv;
        if (ACT == 1) v = v / (1.f + expf(-v));       // SiLU
        Y[(long)(mbase + r) * N + col] = v;
    }
}

// ---------------------------------------------------------------------------
// Zero-pad weight [N,K] -> [Np,Kp] and bias [N] -> [Np]
// ---------------------------------------------------------------------------
__global__ void pad_weight_kernel(const float* __restrict__ W, float* __restrict__ Wp,
                                  int N, int K, int Np, int Kp) {
    int idx = blockIdx.x * blockDim.x + threadIdx.x;
    if (idx >= Np * Kp) return;
    int n = idx / Kp, k = idx % Kp;
    Wp[idx] = (n < N && k < K) ? W[n * K + k] : 0.f;
}

__global__ void pad_bias_kernel(const float* __restrict__ b, float* __restrict__ bp,
                                int N, int Np) {
    int idx = blockIdx.x * blockDim.x + threadIdx.x;
    if (idx >= Np) return;
    bp[idx] = (idx < N) ? b[idx] : 0.f;
}

// ---------------------------------------------------------------------------
// x = h * emb[batch]          (elementwise, 2048x128)
// ---------------------------------------------------------------------------
__global__ void mul_gather_kernel(const float* __restrict__ h,
                                  const float* __restrict__ emb,
                                  const int*   __restrict__ batch,
                                  float* __restrict__ x) {
    int idx = blockIdx.x * blockDim.x + threadIdx.x;
    if (idx >= NN * HID) return;
    int i = idx >> 7;           // node
    int c = idx & 127;          // feature
    x[idx] = h[idx] * emb[batch[i] * HID + c];
}

// ---------------------------------------------------------------------------
// Per-row scalars: b_bias[i] = x[i]·W_b + b_b ;  gate[i] = sigmoid(x[i]·W_g + b_g)
// One wave per row.
// ---------------------------------------------------------------------------
__global__ void row_scalars_kernel(const float* __restrict__ x,
                                   const float* __restrict__ Wb, const float* __restrict__ bb,
                                   const float* __restrict__ Wg, const float* __restrict__ bg,
                                   float* __restrict__ bbias, float* __restrict__ gate) {
    int row  = blockIdx.x * 8 + (threadIdx.x >> 5);
    int lane = threadIdx.x & 31;
    if (row >= NN) return;
    float sb = 0.f, sg = 0.f;
    const float* xr = x + (long)row * HID;
    #pragma unroll
    for (int k = lane; k < HID; k += 32) {
        float xv = xr[k];
        sb += xv * Wb[k];
        sg += xv * Wg[k];
    }
    #pragma unroll
    for (int off = 16; off > 0; off >>= 1) {
        sb += __shfl_xor(sb, off, 32);
        sg += __shfl_xor(sg, off, 32);
    }
    if (lane == 0) {
        bbias[row] = sb + bb[0];
        gate[row]  = 1.f / (1.f + expf(-(sg + bg[0])));
    }
}

// ---------------------------------------------------------------------------
// Fused per-graph attention.  One block (256 thr, 8 waves) per graph.
//   t_ij   : 32-bin gaussian soft-one-hot of pairwise distance, projected
//   S      : q k^T via WMMA (32x32, K=128)     -> logits = S/sqrt(32)+bbias+t
//   softmax over rows (block is fully intra-graph; mask is implicit)
//   O      : attn v via WMMA (32x128, K=32) ;  vals = O*gate + x
// ---------------------------------------------------------------------------
__global__ void attention_kernel(const float* __restrict__ qkv,   // [NN,384]
                                 const float* __restrict__ x,     // [NN,128]
                                 const float* __restrict__ pos,   // [NN,3]
                                 const float* __restrict__ Wt,    // [32]
                                 const float* __restrict__ bt,    // [1]
                                 const float* __restrict__ bbias, // [NN]
                                 const float* __restrict__ gate,  // [NN]
                                 float* __restrict__ vals) {      // [NN,128]
    __shared__ float logits[ATOMS][ATOMS + 1];
    __shared__ float att[ATOMS][ATOMS + 1];

    const int g    = blockIdx.x;
    const int base = g * ATOMS;
    const int tid  = threadIdx.x;
    const int wave = tid >> 5;
    const int lane = tid & 31;
    const int khalf = (lane >> 4) * 2;

    // ---- (a) distance soft-one-hot projection -> logits (t term) ----
    const float step = 10.f / (D1BINS - 1);
    for (int e = tid; e < ATOMS * ATOMS; e += 256) {
        int i = e >> 5, j = e & 31;
        float dx = pos[(base + i) * 3 + 0] - pos[(base + j) * 3 + 0];
        float dy = pos[(base + i) * 3 + 1] - pos[(base + j) * 3 + 1];
        float dz = pos[(base + i) * 3 + 2] - pos[(base + j) * 3 + 2];
        float d  = sqrtf(fmaxf(dx * dx + dy * dy + dz * dz, 1e-12f));
        float acc = bt[0];
        #pragma unroll
        for (int kb = 0; kb < D1BINS; ++kb) {
            float diff = d - step * kb;
            acc += expf(-GAMMA * diff * diff) * Wt[kb];
        }
        logits[i][j] = acc;
    }
    __syncthreads();

    // ---- (b) S = q k^T : 4 tiles of 16x16, one per wave 0..3 ----
    if (wave < 4) {
        const int tm = wave >> 1, tn = wave & 1;
        const int arow = base + tm * 16 + (lane & 15);
        const int bcol = base + tn * 16 + (lane & 15);
        v8f acc = {0.f,0.f,0.f,0.f,0.f,0.f,0.f,0.f};
        const float* qr = qkv + (long)arow * 384 + khalf;        // q at offset 0
        const float* kr = qkv + (long)bcol * 384 + HID + khalf;  // k at offset 128
        #pragma unroll 8
        for (int kk = 0; kk < HID; kk += 4) {
            v2f a, b;
            a.x = qr[kk];  a.y = qr[kk + 1];
            b.x = kr[kk];  b.y = kr[kk + 1];
            acc = wmma_f32_16x16x4(a, b, acc);
        }
        const int jloc  = tn * 16 + (lane & 15);
        const int mbase = tm * 16 + 8 * (lane >> 4);
        #pragma unroll
        for (int r = 0; r < 8; ++r) {
            int iloc = mbase + r;
            logits[iloc][jloc] += acc[r] * INV_SQRT_LEN + bbias[base + iloc];
        }
    }
    __syncthreads();

    // ---- (c) row softmax (one thread per row) ----
    if (tid < ATOMS) {
        float mx = -INFINITY;
        for (int j = 0; j < ATOMS; ++j) mx = fmaxf(mx, logits[tid][j]);
        float s = 0.f;
        for (int j = 0; j < ATOMS; ++j) {
            float e = expf(logits[tid][j] - mx);
            att[tid][j] = e;
            s += e;
        }
        float inv = 1.f / s;
        for (int j = 0; j < ATOMS; ++j) att[tid][j] *= inv;
    }
    __syncthreads();

    // ---- (d) O = attn @ v : 2x8 tiles of 16x16, 2 tiles per wave ----
    for (int t = wave; t < 16; t += 8) {
        const int tm = t >> 3, tn = t & 7;
        const int arow = tm * 16 + (lane & 15);
        const int bn   = tn * 16 + (lane & 15);
        v8f acc = {0.f,0.f,0.f,0.f,0.f,0.f,0.f,0.f};
        #pragma unroll
        for (int kk = 0; kk < ATOMS; kk += 4) {
            v2f a, b;
            a.x = att[arow][kk + khalf];
            a.y = att[arow][kk + khalf + 1];
            b.x = qkv[(long)(base + kk + khalf)     * 384 + 2 * HID + bn];  // v at 256
            b.y = qkv[(long)(base + kk + khalf + 1) * 384 + 2 * HID + bn];
            acc = wmma_f32_16x16x4(a, b, acc);
        }
        const int mbase = tm * 16 + 8 * (lane >> 4);
        #pragma unroll
        for (int r = 0; r < 8; ++r) {
            int gi = base + mbase + r;
            vals[(long)gi * HID + bn] = acc[r] * gate[gi] + x[(long)gi * HID + bn];
        }
    }
}

// ---------------------------------------------------------------------------
// log_softmax over last dim (64).  One wave per row; 2 cols per lane.
// ---------------------------------------------------------------------------
__global__ void log_softmax_kernel(const float* __restrict__ y,  // [NN,64]
                                   float* __restrict__ out) {    // [NN,64]
    int row  = blockIdx.x * 8 + (threadIdx.x >> 5);
    int lane = threadIdx.x & 31;
    if (row >= NN) return;
    float v0 = y[(long)row * D2BINS + lane];
    float v1 = y[(long)row * D2BINS + 32 + lane];
    float mx = fmaxf(v0, v1);
    #pragma unroll
    for (int off = 16; off > 0; off >>= 1) mx = fmaxf(mx, __shfl_xor(mx, off, 32));
    float s = expf(v0 - mx) + expf(v1 - mx);
    #pragma unroll
    for (int off = 16; off > 0; off >>= 1) s += __shfl_xor(s, off, 32);
    float lse = mx + logf(s);
    out[(long)row * D2BINS + lane]      = v0 - lse;
    out[(long)row * D2BINS + 32 + lane] = v1 - lse;
}

// ---------------------------------------------------------------------------
// Host launch
// ---------------------------------------------------------------------------
template <int K, int ACT>
static inline void launch_gemm(const float* X, const float* W, const float* bias,
                               float* Y, int M, int N, hipStream_t s) {
    int tiles  = (M >> 4) * (N >> 4);
    int blocks = (tiles + 7) / 8;
    gemm_wmma_kernel<K, ACT><<<blocks, 256, 0, s>>>(X, W, bias, Y, M, N);
}

extern "C" void kernel_launch(void* const* d_in, const int* in_sizes, int n_in,
                              void* d_out, int out_size, void* d_ws, size_t ws_size,
                              hipStream_t stream) {
    // setup_inputs() order:
    const float* h         = (const float*)d_in[0];   // [2048,128]
    const float* pos       = (const float*)d_in[1];   // [2048,3]
    const float* next_type = (const float*)d_in[2];   // [64,64]
    const int*   batch     = (const int*)  d_in[3];   // [2048]
    const float* W_emb     = (const float*)d_in[4];   // [128,64]
    const float* W_qkv     = (const float*)d_in[5];   // [384,128]
    const float* b_qkv     = (const float*)d_in[6];   // [384]
    const float* W_b       = (const float*)d_in[7];   // [128]
    const float* b_b       = (const float*)d_in[8];   // [1]
    const float* W_g       = (const float*)d_in[9];   // [128]
    const float* b_g       = (const float*)d_in[10];  // [1]
    const float* W_t       = (const float*)d_in[11];  // [32]
    const float* b_t       = (const float*)d_in[12];  // [1]
    const float* W_d0      = (const float*)d_in[13];  // [106,128]
    const float* b_d0      = (const float*)d_in[14];  // [106]
    const float* W_d1      = (const float*)d_in[15];  // [85,106]
    const float* b_d1      = (const float*)d_in[16];  // [85]
    const float* W_d2      = (const float*)d_in[17];  // [64,85]
    const float* b_d2      = (const float*)d_in[18];  // [64]
    float* out = (float*)d_out;

    // workspace carve-up (floats); total ~= 7.6 MB
    float* ws    = (float*)d_ws;
    float* emb   = ws;                      // 64*128      = 8192
    float* x     = emb   + NG * HID;        // 2048*128    = 262144
    float* qkv   = x     + NN * HID;        // 2048*384    = 786432
    float* bbias = qkv   + NN * 3 * HID;    // 2048
    float* gate  = bbias + NN;              // 2048
    float* vals  = gate  + NN;              // 2048*128    = 262144
    float* y0    = vals  + NN * HID;        // 2048*112    = 229376  (padded DIM1)
    float* y1    = y0    + NN * DIM1P;      // 2048*96     = 196608  (padded DIM2)
    float* y2    = y1    + NN * DIM2P;      // 2048*64     = 131072
    float* Wp0   = y2    + NN * D2BINS;     // 112*128     = 14336
    float* bp0   = Wp0   + DIM1P * HID;     // 112
    float* Wp1   = bp0   + DIM1P;           // 96*112      = 10752
    float* bp1   = Wp1   + DIM2P * DIM1P;   // 96
    float* Wp2   = bp1   + DIM2P;           // 64*96       = 6144

    // 0) zero-pad dense-head weights/biases into tile-friendly shapes.
    //    Zero pad weights + zero pad bias => pad output cols are silu(0)=0,
    //    which is exactly the zero K-padding the next layer requires.
    pad_weight_kernel<<<(DIM1P * HID   + 255) / 256, 256, 0, stream>>>(W_d0, Wp0, DIM1, HID,  DIM1P, HID);
    pad_bias_kernel  <<<1, 256, 0, stream>>>(b_d0, bp0, DIM1, DIM1P);
    pad_weight_kernel<<<(DIM2P * DIM1P + 255) / 256, 256, 0, stream>>>(W_d1, Wp1, DIM2, DIM1, DIM2P, DIM1P);
    pad_bias_kernel  <<<1, 256, 0, stream>>>(b_d1, bp1, DIM2, DIM2P);
    pad_weight_kernel<<<(D2BINS * DIM2P + 255) / 256, 256, 0, stream>>>(W_d2, Wp2, D2BINS, DIM2, D2BINS, DIM2P);

    // 1) emb = next_type @ W_emb^T        (64x128, K=64, no bias)
    launch_gemm<FEAT, 0>(next_type, W_emb, nullptr, emb, NG, HID, stream);

    // 2) x = h * emb[batch]
    mul_gather_kernel<<<(NN * HID + 255) / 256, 256, 0, stream>>>(h, emb, batch, x);

    // 3) qkv = x @ W_qkv^T + b_qkv        (2048x384, K=128)
    launch_gemm<HID, 0>(x, W_qkv, b_qkv, qkv, NN, 3 * HID, stream);

    // 4) per-row bias + gate
    row_scalars_kernel<<<NN / 8, 256, 0, stream>>>(x, W_b, b_b, W_g, b_g, bbias, gate);

    // 5) fused per-graph attention -> vals
    attention_kernel<<<NG, 256, 0, stream>>>(qkv, x, pos, W_t, b_t, bbias, gate, vals);

    // 6) dense head (padded): SiLU(vals@Wp0^T+bp0) -> y0[,112],
    //    SiLU(y0@Wp1^T+bp1) -> y1[,96],  y1@Wp2^T+b_d2 -> y2[,64]
    launch_gemm<HID,   1>(vals, Wp0, bp0,  y0, NN, DIM1P,  stream);
    launch_gemm<DIM1P, 1>(y0,   Wp1, bp1,  y1, NN, DIM2P,  stream);
    launch_gemm<DIM2P, 0>(y1,   Wp2, b_d2, y2, NN, D2BINS, stream);

    // 7) log_softmax -> out
    log_softmax_kernel<<<NN / 8, 256, 0, stream>>>(y2, out);

    (void)in_sizes; (void)n_in; (void)out_size; (void)ws_size;
}